// SRModel_57123065037230
// MI455X (gfx1250) — compile-verified
//
#include <hip/hip_runtime.h>
#include <hip/hip_bf16.h>

typedef float v2f __attribute__((ext_vector_type(2)));
typedef float v8f __attribute__((ext_vector_type(8)));
typedef int   v4i __attribute__((vector_size(16)));   // matches builtin param type

#define EPSF 1e-5f

#if defined(__has_builtin)
#if __has_builtin(__builtin_amdgcn_global_load_async_to_lds_b128)
#define HAS_ASYNC_LDS 1
#endif
#endif
#ifndef HAS_ASYNC_LDS
#define HAS_ASYNC_LDS 0
#endif

#if HAS_ASYNC_LDS
#define AS_GLOBAL __attribute__((address_space(1)))
#define AS_LOCAL  __attribute__((address_space(3)))
__device__ __forceinline__ void async_wait0() {
#if __has_builtin(__builtin_amdgcn_s_wait_asynccnt)
    __builtin_amdgcn_s_wait_asynccnt(0);
#else
    asm volatile("s_wait_asynccnt 0x0" ::: "memory");
#endif
}
#endif

// ---------------------------------------------------------------------------
// conv0: x(16,1,128,2048) -> out(16,32,64,1024), 3x3 stride2 pad1, +bias
// ---------------------------------------------------------------------------
__global__ void conv0_k(const float* __restrict__ x, const float* __restrict__ w,
                        const float* __restrict__ bias, float* __restrict__ out) {
    int idx = blockIdx.x * blockDim.x + threadIdx.x;   // 33,554,432 total
    int t  = idx & 1023;
    int f  = (idx >> 10) & 63;
    int co = (idx >> 16) & 31;
    int n  = idx >> 21;
    float acc = bias[co];
    int fi0 = f * 2 - 1, ti0 = t * 2 - 1;
#pragma unroll
    for (int kf = 0; kf < 3; ++kf) {
        int fi = fi0 + kf;
        if (fi < 0 || fi >= 128) continue;
#pragma unroll
        for (int kt = 0; kt < 3; ++kt) {
            int ti = ti0 + kt;
            if (ti < 0 || ti >= 2048) continue;
            acc += w[co * 9 + kf * 3 + kt] * x[(n * 128 + fi) * 2048 + ti];
        }
    }
    out[idx] = acc;
}

// ---------------------------------------------------------------------------
// Residual-block conv: 3x3, 32->32, pad1, + conv bias, BN, (optional +res), ReLU
// Block: 256 threads, tile = 8(f) x 32(t), loops all 32 output channels.
// LDS: input patch [32ci][10f][34t] = 43.5 KB
// ---------------------------------------------------------------------------
__global__ void resconv_k(const float* __restrict__ in, const float* __restrict__ w,
                          const float* __restrict__ cb,
                          const float* __restrict__ bg, const float* __restrict__ bb,
                          const float* __restrict__ bm, const float* __restrict__ bv,
                          const float* __restrict__ resid, float* __restrict__ out,
                          int add_res) {
    __shared__ float sh[32 * 10 * 34];
    int n  = blockIdx.z;
    int f0 = blockIdx.y * 8;
    int t0 = blockIdx.x * 32;
    const float* inN = in + (size_t)n * 32 * 64 * 1024;
    for (int i = threadIdx.x; i < 32 * 10 * 34; i += 256) {
        int ci = i / 340;
        int r  = i - ci * 340;
        int lf = r / 34;
        int lt = r - lf * 34;
        int fi = f0 - 1 + lf;
        int ti = t0 - 1 + lt;
        float v = 0.f;
        if (fi >= 0 && fi < 64 && ti >= 0 && ti < 1024)
            v = inN[(ci * 64 + fi) * 1024 + ti];
        sh[i] = v;
    }
    __syncthreads();
    int lf = threadIdx.x >> 5;   // 0..7
    int lt = threadIdx.x & 31;   // 0..31
    int f = f0 + lf, t = t0 + lt;
    for (int co = 0; co < 32; ++co) {
        float acc = 0.f;
        const float* wc = w + co * 32 * 9;
        for (int ci = 0; ci < 32; ++ci) {
            const float* s  = sh + ci * 340 + lf * 34 + lt;
            const float* wk = wc + ci * 9;
#pragma unroll
            for (int kf = 0; kf < 3; ++kf)
#pragma unroll
                for (int kt = 0; kt < 3; ++kt)
                    acc += s[kf * 34 + kt] * wk[kf * 3 + kt];
        }
        float inv = rsqrtf(bv[co] + EPSF);
        float sc  = bg[co] * inv;
        float y   = (acc + cb[co] - bm[co]) * sc + bb[co];
        int oidx  = ((n * 32 + co) * 64 + f) * 1024 + t;
        if (add_res) y += resid[oidx];
        out[oidx] = fmaxf(y, 0.f);
    }
}

// ---------------------------------------------------------------------------
// LayerNorm over D (<=2048) columns; one block per row (row = t*16 + b).
// gather==1: read from h(16,32,64,1024) with s[t,b,c] = h[(b*2048+c)*1024+t]
// gather==0: read row-major src[row*D + c]. Output row-major dst[row*D + c].
// ---------------------------------------------------------------------------
__global__ void ln_k(const float* __restrict__ src, float* __restrict__ dst,
                     const float* __restrict__ g, const float* __restrict__ b,
                     int D, int gather) {
    __shared__ float sh[2048];
    __shared__ float red[256];
    int row = blockIdx.x;
    int tid = threadIdx.x;
    int tt  = row >> 4;
    int bb_ = row & 15;
    float s1 = 0.f;
    for (int c = tid; c < D; c += 256) {
        float v = gather ? src[((size_t)bb_ * 2048 + c) * 1024 + tt]
                         : src[(size_t)row * D + c];
        sh[c] = v;
        s1 += v;
    }
    red[tid] = s1; __syncthreads();
    for (int s = 128; s > 0; s >>= 1) { if (tid < s) red[tid] += red[tid + s]; __syncthreads(); }
    float mean = red[0] / (float)D;
    __syncthreads();
    float s2 = 0.f;
    for (int c = tid; c < D; c += 256) { float d = sh[c] - mean; s2 += d * d; }
    red[tid] = s2; __syncthreads();
    for (int s = 128; s > 0; s >>= 1) { if (tid < s) red[tid] += red[tid + s]; __syncthreads(); }
    float inv = rsqrtf(red[0] / (float)D + EPSF);
    for (int c = tid; c < D; c += 256)
        dst[(size_t)row * D + c] = (sh[c] - mean) * inv * g[c] + b[c];
}

// ---------------------------------------------------------------------------
// GEMM: C(MxN) = A(MxK) * B(KxN), row-major fp32, via V_WMMA_F32_16X16X4_F32.
// Block = 256 threads = 8 waves (2m x 4n); block tile 64x128; each wave owns a
// 32x32 patch = 4 independent 16x16 accumulators (4 parallel WMMA dep chains).
// K staged in chunks of 16 through double-buffered LDS; next chunk is brought
// in with GLOBAL_LOAD_ASYNC_TO_LDS_B128 (ASYNCcnt) while WMMAs consume the
// current chunk. Requires M%64==0, N%128==0, K%16==0 (true at all call sites).
// ---------------------------------------------------------------------------
__global__ void gemm_k(const float* __restrict__ A, const float* __restrict__ B,
                       float* __restrict__ C, int M, int N, int K) {
    __shared__ __align__(16) float sA[2][64 * 16];    //  8 KB
    __shared__ __align__(16) float sB[2][16 * 128];   // 16 KB
    int tid  = threadIdx.x;
    int wave = tid >> 5;
    int lane = tid & 31;
    int wm32 = (wave >> 2) * 32;  // 0 or 32
    int wn32 = (wave & 3) * 32;   // 0,32,64,96
    int m0 = blockIdx.y * 64;
    int n0 = blockIdx.x * 128;
    int l16   = lane & 15;
    int lhalf = lane >> 4;        // 0 or 1
    int kh    = lhalf * 2;        // f32 frag layout: VGPR v, lane-half h -> K = 2h+v

    v8f acc00 = {}, acc01 = {}, acc10 = {}, acc11 = {};

    // stage one K-chunk (16 wide) into LDS buffer `buf`
    auto stage = [&](int buf, int k0) {
#if HAS_ASYNC_LDS
        {   // A tile 64x16: 256 x 16B segments, one per thread
            int r = tid >> 2, c = (tid & 3) * 4;
            const float* gp = A + (size_t)(m0 + r) * K + k0 + c;
            __builtin_amdgcn_global_load_async_to_lds_b128(
                (AS_GLOBAL v4i*)gp, (AS_LOCAL v4i*)&sA[buf][r * 16 + c], 0, 0);
        }
#pragma unroll
        for (int j = 0; j < 2; ++j) {   // B tile 16x128: 512 segments, two per thread
            int i = tid + j * 256;
            int r = i >> 5, c = (i & 31) * 4;
            const float* gp = B + (size_t)(k0 + r) * N + n0 + c;
            __builtin_amdgcn_global_load_async_to_lds_b128(
                (AS_GLOBAL v4i*)gp, (AS_LOCAL v4i*)&sB[buf][r * 128 + c], 0, 0);
        }
#else
        for (int i = tid; i < 64 * 16; i += 256) {
            int r = i >> 4, c = i & 15;
            sA[buf][i] = A[(size_t)(m0 + r) * K + k0 + c];
        }
        for (int i = tid; i < 16 * 128; i += 256) {
            int r = i >> 7, c = i & 127;
            sB[buf][i] = B[(size_t)(k0 + r) * N + n0 + c];
        }
#endif
    };

    stage(0, 0);
    int p = 0;
    for (int k0 = 0; k0 < K; k0 += 16) {
#if HAS_ASYNC_LDS
        async_wait0();                 // my async writes into buf[p] done
#endif
        __syncthreads();               // all waves: buf[p] ready, buf[p^1] free
        if (k0 + 16 < K) stage(p ^ 1, k0 + 16);
#pragma unroll
        for (int kk = 0; kk < 16; kk += 4) {
            int kb = kk + kh;
            v2f a0, a1, b0, b1;
            a0.x = sA[p][(wm32      + l16) * 16 + kb];
            a0.y = sA[p][(wm32      + l16) * 16 + kb + 1];
            a1.x = sA[p][(wm32 + 16 + l16) * 16 + kb];
            a1.y = sA[p][(wm32 + 16 + l16) * 16 + kb + 1];
            b0.x = sB[p][(kb    ) * 128 + wn32      + l16];
            b0.y = sB[p][(kb + 1) * 128 + wn32      + l16];
            b1.x = sB[p][(kb    ) * 128 + wn32 + 16 + l16];
            b1.y = sB[p][(kb + 1) * 128 + wn32 + 16 + l16];
            acc00 = __builtin_amdgcn_wmma_f32_16x16x4_f32(false, a0, false, b0, (short)0, acc00, false, false);
            acc01 = __builtin_amdgcn_wmma_f32_16x16x4_f32(false, a0, false, b1, (short)0, acc01, false, false);
            acc10 = __builtin_amdgcn_wmma_f32_16x16x4_f32(false, a1, false, b0, (short)0, acc10, false, false);
            acc11 = __builtin_amdgcn_wmma_f32_16x16x4_f32(false, a1, false, b1, (short)0, acc11, false, false);
        }
        p ^= 1;
    }

    // D layout: VGPR v, lane l -> row = 8*(l/16)+v, col = l%16
    int crow = m0 + wm32 + lhalf * 8;
    int ccol = n0 + wn32 + l16;
#pragma unroll
    for (int v = 0; v < 8; ++v) {
        C[(size_t)(crow + v     ) * N + ccol     ] = acc00[v];
        C[(size_t)(crow + v     ) * N + ccol + 16] = acc01[v];
        C[(size_t)(crow + v + 16) * N + ccol     ] = acc10[v];
        C[(size_t)(crow + v + 16) * N + ccol + 16] = acc11[v];
    }
}

// ---------------------------------------------------------------------------
// SRU bidirectional scan. U row = t*16+b, col = (d*k + j)*512 + hh.
// 16384 independent sequences (d,b,hh); sequential over L=1024.
// k==4: residual from U slot 3; k==3: residual from xn[row, d*512+hh].
// ---------------------------------------------------------------------------
__device__ __forceinline__ float sigm(float x) { return 1.f / (1.f + __expf(-x)); }

__global__ void sru_k(const float* __restrict__ U, const float* __restrict__ xn,
                      float* __restrict__ out,
                      const float* __restrict__ vc, const float* __restrict__ bias,
                      int k) {
    int tid = blockIdx.x * blockDim.x + threadIdx.x;  // 0..16383
    int hh = tid & 511;
    int b  = (tid >> 9) & 15;
    int d  = tid >> 13;
    const int H = 512;
    int NU = 2 * k * H;
    float vf = vc[(d * 2 + 0) * H + hh];
    float vr = vc[(d * 2 + 1) * H + hh];
    float bf = bias[(d * 2 + 0) * H + hh];
    float br = bias[(d * 2 + 1) * H + hh];
    float c = 0.f;
    for (int step = 0; step < 1024; ++step) {
        int t   = d ? (1023 - step) : step;
        int row = t * 16 + b;
        const float* up = U + (size_t)row * NU + d * k * H;
        float a0 = up[hh];
        float a1 = up[H + hh];
        float a2 = up[2 * H + hh];
        float rx = (k == 4) ? up[3 * H + hh] : xn[(size_t)row * 1024 + d * H + hh];
        float f = sigm(a1 + vf * c + bf);
        c = f * c + (1.f - f) * a0;
        float r = sigm(a2 + vr * c + br);
        out[(size_t)row * 1024 + d * H + hh] = r * c + (1.f - r) * rx;
    }
}

// ---------------------------------------------------------------------------
// Final: per (b,t): LayerNorm(1024) then 30-way classifier.
// ---------------------------------------------------------------------------
__global__ void cls_k(const float* __restrict__ s, const float* __restrict__ g,
                      const float* __restrict__ b, const float* __restrict__ W,
                      float* __restrict__ out) {
    __shared__ float sh[1024];
    __shared__ float red[256];
    __shared__ float part[30 * 8];
    int t   = blockIdx.x;   // 0..1023
    int bb  = blockIdx.y;   // 0..15
    int row = t * 16 + bb;
    int tid = threadIdx.x;
    float s1 = 0.f;
    for (int c = tid; c < 1024; c += 256) {
        float v = s[(size_t)row * 1024 + c];
        sh[c] = v; s1 += v;
    }
    red[tid] = s1; __syncthreads();
    for (int st = 128; st > 0; st >>= 1) { if (tid < st) red[tid] += red[tid + st]; __syncthreads(); }
    float mean = red[0] * (1.f / 1024.f);
    __syncthreads();
    float s2 = 0.f;
    for (int c = tid; c < 1024; c += 256) { float d = sh[c] - mean; s2 += d * d; }
    red[tid] = s2; __syncthreads();
    for (int st = 128; st > 0; st >>= 1) { if (tid < st) red[tid] += red[tid + st]; __syncthreads(); }
    float inv = rsqrtf(red[0] * (1.f / 1024.f) + EPSF);
    for (int c = tid; c < 1024; c += 256)
        sh[c] = (sh[c] - mean) * inv * g[c] + b[c];
    __syncthreads();
    int cls = tid >> 3, seg = tid & 7;
    if (cls < 30) {
        float p = 0.f;
        for (int i = 0; i < 128; ++i) {
            int c = seg * 128 + i;
            p += sh[c] * W[c * 30 + cls];
        }
        part[cls * 8 + seg] = p;
    }
    __syncthreads();
    if (tid < 30) {
        float acc = 0.f;
#pragma unroll
        for (int i = 0; i < 8; ++i) acc += part[tid * 8 + i];
        out[((size_t)bb * 1024 + t) * 30 + tid] = acc;
    }
}

// ---------------------------------------------------------------------------
extern "C" void kernel_launch(void* const* d_in, const int* in_sizes, int n_in,
                              void* d_out, int out_size, void* d_ws, size_t ws_size,
                              hipStream_t stream) {
    const float* x        = (const float*)d_in[0];
    const float* conv0_w  = (const float*)d_in[1];
    const float* conv0_b  = (const float*)d_in[2];
    const float* rconv1_w = (const float*)d_in[3];
    const float* rconv1_b = (const float*)d_in[4];
    const float* rbn1_g   = (const float*)d_in[5];
    const float* rbn1_b   = (const float*)d_in[6];
    const float* rbn1_m   = (const float*)d_in[7];
    const float* rbn1_v   = (const float*)d_in[8];
    const float* rconv2_w = (const float*)d_in[9];
    const float* rconv2_b = (const float*)d_in[10];
    const float* rbn2_g   = (const float*)d_in[11];
    const float* rbn2_b   = (const float*)d_in[12];
    const float* rbn2_m   = (const float*)d_in[13];
    const float* rbn2_v   = (const float*)d_in[14];
    const float* ln0_g    = (const float*)d_in[15];
    const float* ln0_b    = (const float*)d_in[16];
    const float* wproj0   = (const float*)d_in[17];
    const float* w0       = (const float*)d_in[18];
    const float* vc0      = (const float*)d_in[19];
    const float* bias0    = (const float*)d_in[20];
    const float* ln_g     = (const float*)d_in[21];
    const float* ln_b     = (const float*)d_in[22];
    const float* wproj    = (const float*)d_in[23];
    const float* wmat     = (const float*)d_in[24];
    const float* vc       = (const float*)d_in[25];
    const float* biasv    = (const float*)d_in[26];
    const float* cln_g    = (const float*)d_in[27];
    const float* cln_b    = (const float*)d_in[28];
    const float* cls_w    = (const float*)d_in[29];

    // Workspace layout (f32 elements)
    float* hA = (float*)d_ws;                  // 16*32*64*1024 = 33,554,432
    float* hB = hA + 33554432u;                // 33,554,432
    float* U  = hB + 33554432u;                // 16384*4096    = 67,108,864
    float* xn = U  + 67108864u;                // 16384*2048    = 33,554,432
    float* pr = xn + 33554432u;                // 16384*256     =  4,194,304
    float* sA = pr + 4194304u;                 // 16384*1024    = 16,777,216
    float* sB = sA + 16777216u;                // 16,777,216

    // 1) conv0
    conv0_k<<<131072, 256, 0, stream>>>(x, conv0_w, conv0_b, hA);

    // 2) 3 residual blocks
    dim3 cg(32, 8, 16);  // t-tiles, f-tiles, batch
    for (int i = 0; i < 3; ++i) {
        resconv_k<<<cg, 256, 0, stream>>>(hA, rconv1_w + i * 9216, rconv1_b + i * 32,
                                          rbn1_g + i * 32, rbn1_b + i * 32,
                                          rbn1_m + i * 32, rbn1_v + i * 32,
                                          hA /*unused*/, hB, 0);
        resconv_k<<<cg, 256, 0, stream>>>(hB, rconv2_w + i * 9216, rconv2_b + i * 32,
                                          rbn2_g + i * 32, rbn2_b + i * 32,
                                          rbn2_m + i * 32, rbn2_v + i * 32,
                                          hA, hA, 1);   // in-place pointwise residual
    }

    // 3) SRU layer 0 (D=2048, k=4)
    ln_k<<<16384, 256, 0, stream>>>(hA, xn, ln0_g, ln0_b, 2048, 1);
    gemm_k<<<dim3(256 / 128, 16384 / 64), 256, 0, stream>>>(xn, wproj0, pr, 16384, 256, 2048);
    gemm_k<<<dim3(4096 / 128, 16384 / 64), 256, 0, stream>>>(pr, w0, U, 16384, 4096, 256);
    sru_k<<<64, 256, 0, stream>>>(U, xn, sA, vc0, bias0, 4);

    // 4) SRU layers 1..3 (D=1024, k=3)
    float* cur = sA;
    float* nxt = sB;
    for (int i = 0; i < 3; ++i) {
        ln_k<<<16384, 256, 0, stream>>>(cur, xn, ln_g + i * 1024, ln_b + i * 1024, 1024, 0);
        gemm_k<<<dim3(256 / 128, 16384 / 64), 256, 0, stream>>>(xn, wproj + i * 1024 * 256,
                                                                pr, 16384, 256, 1024);
        gemm_k<<<dim3(3072 / 128, 16384 / 64), 256, 0, stream>>>(pr, wmat + i * 256 * 3072,
                                                                 U, 16384, 3072, 256);
        sru_k<<<64, 256, 0, stream>>>(U, xn, nxt, vc + i * 2048, biasv + i * 2048, 3);
        float* tmp = cur; cur = nxt; nxt = tmp;
    }

    // 5) final LN + classifier
    cls_k<<<dim3(1024, 16), 256, 0, stream>>>(cur, cln_g, cln_b, cls_w, (float*)d_out);
}